// GIoULoss_19808389169788
// MI455X (gfx1250) — compile-verified
//
#include <hip/hip_runtime.h>

typedef __attribute__((ext_vector_type(4))) float f4;
typedef __attribute__((ext_vector_type(2))) float v2f;
typedef __attribute__((ext_vector_type(8))) float v8f;

#define GIOU_EPS 1e-7f
#define TPB 256
#define NBLOCKS 2048

__device__ __forceinline__ float giou3d(
    float px1, float py1, float px2, float py2, float pz1, float pz2,
    float tx1, float ty1, float tx2, float ty2, float tz1, float tz2)
{
    float vol1 = (px2 - px1) * (py2 - py1) * (pz2 - pz1);
    float vol2 = (tx2 - tx1) * (ty2 - ty1) * (tz2 - tz1);
    float ix = fmaxf(fminf(px2, tx2) - fmaxf(px1, tx1), 0.0f);
    float iy = fmaxf(fminf(py2, ty2) - fmaxf(py1, ty1), 0.0f);
    float iz = fmaxf(fminf(pz2, tz2) - fmaxf(pz1, tz1), 0.0f);
    float inter = ix * iy * iz;
    float uni   = vol1 + vol2 - inter;
    float iou   = inter / uni;
    float ex = fmaxf(fmaxf(px2, tx2) - fminf(px1, tx1), 0.0f);
    float ey = fmaxf(fmaxf(py2, ty2) - fminf(py1, ty1), 0.0f);
    float ez = fmaxf(fmaxf(pz2, tz2) - fminf(pz1, tz1), 0.0f);
    float enc = ex * ey * ez + GIOU_EPS;
    return iou - (enc - uni) / enc;
}

// Pass 1: each thread streams pairs of boxes (3x b128 NT loads per array),
// wave-reduces its partial via WMMA (ones(16x4) x B -> column sums), block
// combines 8 wave results in fixed order, writes one float per block.
__global__ void giou_partial_kernel(const float* __restrict__ pred,
                                    const float* __restrict__ targ,
                                    float* __restrict__ block_out,
                                    long long npairs, long long nboxes)
{
    float acc = 0.0f;
    const long long stride = (long long)gridDim.x * blockDim.x;
    for (long long i = (long long)blockIdx.x * blockDim.x + threadIdx.x;
         i < npairs; i += stride) {
        const f4* p4 = (const f4*)(pred + 12LL * i);   // 48B-aligned: 16B ok
        const f4* t4 = (const f4*)(targ + 12LL * i);
        f4 p0 = __builtin_nontemporal_load(p4 + 0);
        f4 p1 = __builtin_nontemporal_load(p4 + 1);
        f4 p2 = __builtin_nontemporal_load(p4 + 2);
        f4 t0 = __builtin_nontemporal_load(t4 + 0);
        f4 t1 = __builtin_nontemporal_load(t4 + 1);
        f4 t2 = __builtin_nontemporal_load(t4 + 2);
        // box 2i:   [p0.x p0.y p0.z p0.w p1.x p1.y]
        acc += giou3d(p0.x, p0.y, p0.z, p0.w, p1.x, p1.y,
                      t0.x, t0.y, t0.z, t0.w, t1.x, t1.y);
        // box 2i+1: [p1.z p1.w p2.x p2.y p2.z p2.w]
        acc += giou3d(p1.z, p1.w, p2.x, p2.y, p2.z, p2.w,
                      t1.z, t1.w, t2.x, t2.y, t2.z, t2.w);
    }
    // Odd tail box (generic-N safety; N=4M is even so this is dead here)
    if (blockIdx.x == 0 && threadIdx.x == 0 && (nboxes & 1LL)) {
        const float* pb = pred + 6LL * (nboxes - 1);
        const float* tb = targ + 6LL * (nboxes - 1);
        acc += giou3d(pb[0], pb[1], pb[2], pb[3], pb[4], pb[5],
                      tb[0], tb[1], tb[2], tb[3], tb[4], tb[5]);
    }

    // ---- wave32 reduction ----
    float wsum;
#if __has_builtin(__builtin_amdgcn_wmma_f32_16x16x4_f32)
    {
        // D = ones(16x4) * B + 0 ; B's 2 VGPRs x 32 lanes map bijectively
        // onto the 4x16 matrix, so row 0 of D (c[0], lanes 0..15) holds 16
        // column sums whose total is exactly the 32-lane sum (f32-exact, x1.0).
        v2f a; a.x = 1.0f; a.y = 1.0f;
        v2f b; b.x = acc;  b.y = 0.0f;
        v8f c = {};
        c = __builtin_amdgcn_wmma_f32_16x16x4_f32(
                /*neg_a=*/false, a, /*neg_b=*/false, b,
                /*c_mod=*/(short)0, c, /*reuse_a=*/false, /*reuse_b=*/false);
        wsum = c[0];
        wsum += __shfl_xor(wsum, 1, 32);
        wsum += __shfl_xor(wsum, 2, 32);
        wsum += __shfl_xor(wsum, 4, 32);
        wsum += __shfl_xor(wsum, 8, 32);   // lane 0: sum of 16 column sums
    }
#else
    wsum = acc;
    for (int m = 16; m >= 1; m >>= 1) wsum += __shfl_xor(wsum, m, 32);
#endif

    __shared__ float wpart[TPB / 32];
    const int lane = threadIdx.x & 31;
    const int wid  = threadIdx.x >> 5;
    if (lane == 0) wpart[wid] = wsum;
    __syncthreads();
    if (threadIdx.x == 0) {
        float bsum = 0.0f;
        #pragma unroll
        for (int w = 0; w < TPB / 32; ++w) bsum += wpart[w];  // fixed order
        block_out[blockIdx.x] = bsum;
    }
}

// Pass 2: one block, fixed-order strided accumulate + deterministic LDS tree.
__global__ void giou_final_kernel(const float* __restrict__ block_out,
                                  int nblocks, float* __restrict__ out)
{
    __shared__ float s[TPB];
    float v = 0.0f;
    for (int i = threadIdx.x; i < nblocks; i += TPB) v += block_out[i];
    s[threadIdx.x] = v;
    __syncthreads();
    for (int off = TPB / 2; off > 0; off >>= 1) {
        if ((int)threadIdx.x < off) s[threadIdx.x] += s[threadIdx.x + off];
        __syncthreads();
    }
    if (threadIdx.x == 0) out[0] = -s[0];   // LOSS_WEIGHT * -1 * sum
}

extern "C" void kernel_launch(void* const* d_in, const int* in_sizes, int n_in,
                              void* d_out, int out_size, void* d_ws, size_t ws_size,
                              hipStream_t stream) {
    const float* pred = (const float*)d_in[0];
    const float* targ = (const float*)d_in[1];
    const long long nboxes = (long long)in_sizes[0] / 6;
    const long long npairs = nboxes / 2;

    int blocks = NBLOCKS;
    if ((size_t)blocks * sizeof(float) > ws_size)
        blocks = (int)(ws_size / sizeof(float));
    if (blocks < 1) blocks = 1;

    float* bpart = (float*)d_ws;
    giou_partial_kernel<<<blocks, TPB, 0, stream>>>(pred, targ, bpart,
                                                    npairs, nboxes);
    giou_final_kernel<<<1, TPB, 0, stream>>>(bpart, blocks, (float*)d_out);
}